// GraphConvolution_87694642250199
// MI455X (gfx1250) — compile-verified
//
#include <hip/hip_runtime.h>
#include <hip/hip_bf16.h>

#define N_NODES 100000
#define N_EDGES 1600000
#define IN_DIM  128
#define OUT_DIM 64

typedef float v2f __attribute__((ext_vector_type(2)));
typedef float v8f __attribute__((ext_vector_type(8)));

typedef unsigned int u32x4 __attribute__((ext_vector_type(4)));
typedef int          i32x4 __attribute__((ext_vector_type(4)));
typedef int          i32x8 __attribute__((ext_vector_type(8)));

#if defined(__gfx1250__) && __has_builtin(__builtin_amdgcn_tensor_load_to_lds) && \
    __has_builtin(__builtin_amdgcn_s_wait_tensorcnt)
#define USE_TDM 1
#else
#define USE_TDM 0
#endif

// ---------------------------------------------------------------------------
// Kernel 1: support = x @ W  using V_WMMA_F32_16X16X4_F32 (exact f32 GEMM)
// Block = 128 threads (4 wave32). Block owns one 16-row M-tile; wave w owns
// output columns [16w, 16w+16). A-tile (16x128) and W (128x64) staged in LDS
// with padded strides for conflict-free ds reads. Staging uses the Tensor
// Data Mover (D# pad fields produce the padded layout directly) when the
// builtin is available; manual float4 staging otherwise.
// ---------------------------------------------------------------------------
#define SA_STRIDE 132   // 16x128 A tile: pad 4 DWORDs every 128 (528B rows, 16B aligned)
#define SW_STRIDE 72    // 128x64 W:      pad 8 DWORDs every 64  (288B rows, 16B aligned)

#if USE_TDM
// Issue one 2-D TDM tile load: global (row-major, stride0 elements/row) -> LDS,
// with pad_amount DWORDs of LDS padding inserted every pad_interval DWORDs.
// D# layout per CDNA5 ISA 8.3/8.4. data_size = 4 bytes. 2-D: groups 2/3 zero.
__device__ __forceinline__ void tdm_load_2d_f32(
    const void* gaddr, unsigned lds_addr,
    unsigned tensor_d0, unsigned tensor_d1,
    unsigned tile_d0, unsigned tile_d1,
    unsigned stride0,
    unsigned pad_interval_code,   // interval = 2^(code+1) DWORDs
    unsigned pad_amount_code) {   // amount  = code+1 DWORDs
  const unsigned long long ga = (unsigned long long)gaddr;
  u32x4 g0;
  g0[0] = 1u;                                            // count=1, is_restore=0
  g0[1] = lds_addr;                                      // bits 63:32  lds_addr
  g0[2] = (unsigned)(ga & 0xFFFFFFFFu);                  // bits 95:64  global_addr lo
  g0[3] = (unsigned)((ga >> 32) & 0x01FFFFFFu)           // bits 120:96 global_addr hi
        | (2u << 30);                                    // bits 127:126 type=2 (image)
  i32x8 g1;
  g1[0] = (int)((2u << 16)                               // data_size = 4B
              | (1u << 20)                               // pad_enable
              | (pad_interval_code << 22)
              | (pad_amount_code << 25));
  g1[1] = (int)((tensor_d0 & 0xFFFFu) << 16);            // tensor_dim0[15:0]
  g1[2] = (int)((tensor_d0 >> 16) | ((tensor_d1 & 0xFFFFu) << 16));
  g1[3] = (int)((tensor_d1 >> 16) | (tile_d0 << 16));    // tile_dim0
  g1[4] = (int)(tile_d1 & 0xFFFFu);                      // tile_dim1 (tile_dim2=0)
  g1[5] = (int)stride0;                                  // tensor_dim0_stride lo
  g1[6] = 0;                                             // stride hi / dim1_stride lo
  g1[7] = 0;
  const i32x4 z4 = {0, 0, 0, 0};
  const i32x8 z8 = {0, 0, 0, 0, 0, 0, 0, 0};
  // 6-arg toolchain form: (g0, g1, g2, g3, extra, cpol); 2-D -> trailing zero.
  __builtin_amdgcn_tensor_load_to_lds(g0, g1, z4, z4, z8, 0);
}
#endif

__global__ __launch_bounds__(128) void gemm_support_kernel(
    const float* __restrict__ x, const float* __restrict__ w,
    float* __restrict__ support) {
  __shared__ float sA[16 * SA_STRIDE];
  __shared__ float sW[IN_DIM * SW_STRIDE];
  const int tid    = threadIdx.x;
  const int m_base = blockIdx.x * 16;

#if USE_TDM
  if (tid < 32) {  // one wave issues both DMA descriptors (EXEC-independent)
    // W: 128x64, pad 8 DWORDs every 64 -> SW_STRIDE=72  (codes: 5 -> 64, 7 -> 8)
    tdm_load_2d_f32(w, (unsigned)(unsigned long long)(void*)sW,
                    OUT_DIM, IN_DIM, OUT_DIM, IN_DIM, OUT_DIM, 5u, 7u);
    // A tile: 16x128, pad 4 DWORDs every 128 -> SA_STRIDE=132 (codes: 6 -> 128, 3 -> 4)
    tdm_load_2d_f32(x + (size_t)m_base * IN_DIM,
                    (unsigned)(unsigned long long)(void*)sA,
                    IN_DIM, 16, IN_DIM, 16, IN_DIM, 6u, 3u);
    __builtin_amdgcn_s_wait_tensorcnt(0);
  }
  __syncthreads();
#else
  // Manual staging fallback: coalesced float4 loads -> LDS.
  const float4* x4 = (const float4*)(x + (size_t)m_base * IN_DIM);
  #pragma unroll
  for (int f = tid; f < 16 * (IN_DIM / 4); f += 128) {
    int row = f >> 5, c4 = f & 31;
    ((float4*)(sA + row * SA_STRIDE))[c4] = x4[f];
  }
  const float4* w4 = (const float4*)w;
  #pragma unroll
  for (int f = tid; f < IN_DIM * (OUT_DIM / 4); f += 128) {
    int row = f >> 4, c4 = f & 15;
    ((float4*)(sW + row * SW_STRIDE))[c4] = w4[f];
  }
  __syncthreads();
#endif

  const int lane = tid & 31;
  const int wv   = tid >> 5;
  const int nb   = wv * 16;        // N-tile base for this wave
  const int mm   = lane & 15;      // A row within tile
  const int h    = lane >> 4;      // lane half: K = k0 + 2h + j (ISA 7.12.2)
  const int n    = lane & 15;      // B/C/D column within tile

  v8f acc = {0.f, 0.f, 0.f, 0.f, 0.f, 0.f, 0.f, 0.f};
  #pragma unroll
  for (int k0 = 0; k0 < IN_DIM; k0 += 4) {
    const int k = k0 + 2 * h;
    v2f a, b;
    a.x = sA[mm * SA_STRIDE + k];
    a.y = sA[mm * SA_STRIDE + k + 1];
    b.x = sW[k       * SW_STRIDE + nb + n];
    b.y = sW[(k + 1) * SW_STRIDE + nb + n];
    acc = __builtin_amdgcn_wmma_f32_16x16x4_f32(
        /*neg_a=*/false, a, /*neg_b=*/false, b,
        /*c_mod=*/(short)0, acc, /*reuse_a=*/false, /*reuse_b=*/false);
  }

  // C/D layout: VGPR r -> row (r + 8h), lane&15 -> column.
  #pragma unroll
  for (int r = 0; r < 8; ++r) {
    const int row = m_base + r + 8 * h;
    support[(size_t)row * OUT_DIM + nb + n] = acc[r];
  }
}

// ---------------------------------------------------------------------------
// Kernel 2: zero the output accumulator (atomics below accumulate into it).
// ---------------------------------------------------------------------------
__global__ __launch_bounds__(256) void zero_out_kernel(float4* __restrict__ out4) {
  const int i = blockIdx.x * 256 + threadIdx.x;
  out4[i] = make_float4(0.f, 0.f, 0.f, 0.f);
}

// ---------------------------------------------------------------------------
// Kernel 3: edge scatter. 16 lanes per edge, each lane gathers a float4 of
// support[src] (L2-resident: 25.6MB << 192MB L2) and does 4
// global_atomic_add_f32 into out[dst]. Edge arrays are streamed once ->
// nontemporal loads keep them from displacing the hot support set.
// ---------------------------------------------------------------------------
__global__ __launch_bounds__(256) void scatter_kernel(
    const float* __restrict__ support, const float* __restrict__ ew,
    const int* __restrict__ src, const int* __restrict__ dst,
    float* __restrict__ out) {
  const int idx = blockIdx.x * 256 + threadIdx.x;   // < 25.6M
  const int e   = idx >> 4;
  if (e >= N_EDGES) return;
  const int c4  = idx & 15;
  const float wgt = __builtin_nontemporal_load(ew + e);
  const int s = __builtin_nontemporal_load(src + e);
  const int d = __builtin_nontemporal_load(dst + e);
  const float4 v = ((const float4*)support)[(size_t)s * (OUT_DIM / 4) + c4];
  float* o = out + (size_t)d * OUT_DIM + c4 * 4;
  atomicAdd(o + 0, v.x * wgt);
  atomicAdd(o + 1, v.y * wgt);
  atomicAdd(o + 2, v.z * wgt);
  atomicAdd(o + 3, v.w * wgt);
}

// ---------------------------------------------------------------------------
// Kernel 4: out = relu(out + bias), float4-wide. Exact grid.
// ---------------------------------------------------------------------------
__global__ __launch_bounds__(256) void bias_relu_kernel(
    float4* __restrict__ out4, const float* __restrict__ bias) {
  const int i  = blockIdx.x * 256 + threadIdx.x;    // < 1.6M float4
  const int c4 = i & (OUT_DIM / 4 - 1);             // column group 0..15
  const float4 b = ((const float4*)bias)[c4];
  float4 v = out4[i];
  v.x = fmaxf(v.x + b.x, 0.f);
  v.y = fmaxf(v.y + b.y, 0.f);
  v.z = fmaxf(v.z + b.z, 0.f);
  v.w = fmaxf(v.w + b.w, 0.f);
  out4[i] = v;
}

// ---------------------------------------------------------------------------
extern "C" void kernel_launch(void* const* d_in, const int* in_sizes, int n_in,
                              void* d_out, int out_size, void* d_ws, size_t ws_size,
                              hipStream_t stream) {
  const float* x    = (const float*)d_in[0];  // [N_NODES, IN_DIM]
  const float* w    = (const float*)d_in[1];  // [IN_DIM, OUT_DIM]
  const float* bias = (const float*)d_in[2];  // [OUT_DIM]
  const float* ew   = (const float*)d_in[3];  // [N_EDGES]
  const int*   src  = (const int*)d_in[4];    // [N_EDGES]
  const int*   dst  = (const int*)d_in[5];    // [N_EDGES]
  float* out        = (float*)d_out;          // [N_NODES, OUT_DIM]
  float* support    = (float*)d_ws;           // [N_NODES, OUT_DIM] scratch (25.6 MB)

  // 1) support = x @ W   (6250 M-tiles of 16 rows, exact)
  gemm_support_kernel<<<N_NODES / 16, 128, 0, stream>>>(x, w, support);

  // 2) out = 0
  zero_out_kernel<<<(N_NODES * OUT_DIM / 4) / 256, 256, 0, stream>>>((float4*)out);

  // 3) out[dst] += ew * support[src]
  const int scatter_threads = N_EDGES * 16;
  scatter_kernel<<<(scatter_threads + 255) / 256, 256, 0, stream>>>(
      support, ew, src, dst, out);

  // 4) out = relu(out + bias)
  bias_relu_kernel<<<(N_NODES * OUT_DIM / 4) / 256, 256, 0, stream>>>(
      (float4*)out, bias);
}